// Time_feat_query_51436528336960
// MI455X (gfx1250) — compile-verified
//
#include <hip/hip_runtime.h>
#include <hip/hip_bf16.h>
#include <math.h>

typedef __attribute__((ext_vector_type(2))) float v2f;
typedef __attribute__((ext_vector_type(8))) float v8f;

#define HW128 16384   // 128*128
#define HW64  4096
#define HW32  1024

__device__ __forceinline__ int iclamp(int v, int lo, int hi) {
    return v < lo ? lo : (v > hi ? hi : v);
}

// Half-pixel bilinear sample of an SxS plane at output coord (h,w) of a 128x128 grid.
// Edge renormalization of jax.image.resize's linear kernel == index clamping for 2-tap.
__device__ __forceinline__ float bilin(const float* __restrict__ p, int S, int h, int w) {
    float sc = (float)S * (1.0f / 128.0f);
    float y = (h + 0.5f) * sc - 0.5f;
    float x = (w + 0.5f) * sc - 0.5f;
    float yf = floorf(y), xf = floorf(x);
    float fy = y - yf, fx = x - xf;
    int y0 = (int)yf, x0 = (int)xf;
    int y0c = iclamp(y0, 0, S - 1), y1c = iclamp(y0 + 1, 0, S - 1);
    int x0c = iclamp(x0, 0, S - 1), x1c = iclamp(x0 + 1, 0, S - 1);
    float v00 = p[y0c * S + x0c], v01 = p[y0c * S + x1c];
    float v10 = p[y1c * S + x0c], v11 = p[y1c * S + x1c];
    return (1.0f - fy) * ((1.0f - fx) * v00 + fx * v01)
         +          fy * ((1.0f - fx) * v10 + fx * v11);
}

// 4 consecutive horizontal outputs sharing the vertical taps.
__device__ __forceinline__ float4 bilin4(const float* __restrict__ p, int S, int h, int w0) {
    float sc = (float)S * (1.0f / 128.0f);
    float y = (h + 0.5f) * sc - 0.5f;
    float yf = floorf(y);
    float fy = y - yf;
    int y0 = (int)yf;
    int y0c = iclamp(y0, 0, S - 1), y1c = iclamp(y0 + 1, 0, S - 1);
    const float* __restrict__ r0 = p + y0c * S;
    const float* __restrict__ r1 = p + y1c * S;
    float o[4];
#pragma unroll
    for (int j = 0; j < 4; ++j) {
        float x = (w0 + j + 0.5f) * sc - 0.5f;
        float xf = floorf(x);
        float fx = x - xf;
        int x0 = (int)xf;
        int x0c = iclamp(x0, 0, S - 1), x1c = iclamp(x0 + 1, 0, S - 1);
        float v0 = r0[x0c] * (1.0f - fx) + r0[x1c] * fx;
        float v1 = r1[x0c] * (1.0f - fx) + r1[x1c] * fx;
        o[j] = (1.0f - fy) * v0 + fy * v1;
    }
    return make_float4(o[0], o[1], o[2], o[3]);
}

// ---------------- Stage 1: per-pixel channel sums (C=256), 4 pixels/thread ----------------
__global__ void channel_sum4_k(const float* __restrict__ src, float* __restrict__ dst,
                               int C, int HW, int total4) {
    int i = blockIdx.x * blockDim.x + threadIdx.x;
    if (i >= total4) return;
    int HW4 = HW >> 2;
    int b = i / HW4;
    int p4 = (i - b * HW4) << 2;
    const float* s = src + (size_t)b * C * HW + p4;
    float4 acc = make_float4(0.0f, 0.0f, 0.0f, 0.0f);
    for (int c = 0; c < C; ++c) {
        float4 v = *(const float4*)(s + (size_t)c * HW);
        acc.x += v.x; acc.y += v.y; acc.z += v.z; acc.w += v.w;
    }
    *(float4*)(dst + (size_t)i * 4) = acc;
}

// ---------------- Stage 2: mean_feat = (sumL1 + up(sumL2) + up(sumL3)) / 768 ----------------
__global__ void mean_k(const float* __restrict__ s1, const float* __restrict__ s2,
                       const float* __restrict__ s3, float* __restrict__ mean) {
    int i = blockIdx.x * blockDim.x + threadIdx.x;   // 4*128*128
    if (i >= 4 * HW128) return;
    int w = i & 127, h = (i >> 7) & 127, b = i >> 14;
    float v = s1[i] + bilin(s2 + (size_t)b * HW64, 64, h, w)
                    + bilin(s3 + (size_t)b * HW32, 32, h, w);
    mean[i] = v * (1.0f / 768.0f);
}

// ---------------- Stage 3: query_feat with the (Q,B)->(B,Q) reshuffle ----------------
// Row k of query_feat (k = b2*64+q2) draws patch of (batch = k%4, query = k/4).
__global__ void build_qf_k(const float* __restrict__ mean, float* __restrict__ qf) {
    int i = blockIdx.x * blockDim.x + threadIdx.x;   // 256*256
    if (i >= 256 * 256) return;
    int d = i & 255, k = i >> 8;
    int bb = k & 3, q = k >> 2;
    int qi = q >> 3, qj = q & 7;
    int dr = d >> 4, dc = d & 15;
    int r = qi * 16 + dr, cc = qj * 16 + dc;
    qf[i] = mean[(bb << 14) + (r << 7) + cc];
}

// ---------------- Stage 4: GEMM1 via V_WMMA_F32_16X16X4_F32 ----------------
// X: 256x256, W: 256x128 row-major, H = relu(X@W + b), one wave per 16x16 tile.
__global__ void mlp_gemm1_k(const float* __restrict__ X,
                            const float* __restrict__ W1a, const float* __restrict__ B1a,
                            const float* __restrict__ W1b, const float* __restrict__ B1b,
                            float* __restrict__ Ha, float* __restrict__ Hb) {
    const int lane = threadIdx.x;          // blockDim = 32 (one wave32)
    const int m0 = blockIdx.x * 16;
    const int n0 = blockIdx.y * 16;
    const bool snd = blockIdx.z != 0;
    const float* W = snd ? W1b : W1a;
    const float* Bv = snd ? B1b : B1a;
    float* H = snd ? Hb : Ha;
    const int nl = lane & 15;
    const int hi = lane >> 4;
    v8f acc = {};
    for (int k0 = 0; k0 < 256; k0 += 4) {
        int ka = k0 + hi * 2;
        v2f a, b;
        a.x = X[(m0 + nl) * 256 + ka];
        a.y = X[(m0 + nl) * 256 + ka + 1];
        b.x = W[ka * 128 + n0 + nl];
        b.y = W[(ka + 1) * 128 + n0 + nl];
        acc = __builtin_amdgcn_wmma_f32_16x16x4_f32(false, a, false, b,
                                                    (short)0, acc, false, false);
    }
    float bias = Bv[n0 + nl];
#pragma unroll
    for (int r = 0; r < 8; ++r) {
        int m = m0 + r + hi * 8;
        float v = acc[r] + bias;
        H[m * 128 + n0 + nl] = v > 0.0f ? v : 0.0f;
    }
}

// ---------------- Stage 5: GEMM2 (K=128, N padded to 16) ----------------
__global__ void mlp_gemm2_k(const float* __restrict__ Ha, const float* __restrict__ W2a,
                            const float* __restrict__ B2a,
                            const float* __restrict__ Hb, const float* __restrict__ W2b,
                            const float* __restrict__ B2b,
                            float* __restrict__ offv, float* __restrict__ sclv) {
    const int lane = threadIdx.x;          // blockDim = 32
    const int m0 = blockIdx.x * 16;
    const bool sc = blockIdx.y != 0;
    const float* H = sc ? Hb : Ha;
    const float* W = sc ? W2b : W2a;
    const float* Bi = sc ? B2b : B2a;
    const int ncol = sc ? 1 : 2;
    const int nl = lane & 15;
    const int hi = lane >> 4;
    v8f acc = {};
    for (int k0 = 0; k0 < 128; k0 += 4) {
        int ka = k0 + hi * 2;
        v2f a, b;
        a.x = H[(m0 + nl) * 128 + ka];
        a.y = H[(m0 + nl) * 128 + ka + 1];
        b.x = (nl < ncol) ? W[ka * ncol + nl] : 0.0f;
        b.y = (nl < ncol) ? W[(ka + 1) * ncol + nl] : 0.0f;
        acc = __builtin_amdgcn_wmma_f32_16x16x4_f32(false, a, false, b,
                                                    (short)0, acc, false, false);
    }
#pragma unroll
    for (int r = 0; r < 8; ++r) {
        int m = m0 + r + hi * 8;
        if (nl < ncol) {
            float v = acc[r] + Bi[nl];
            if (sc) sclv[m] = v;
            else    offv[m * 2 + nl] = v;
        }
    }
}

// ---------------- Stage 6: per-query rectangles ----------------
__global__ void rects_k(const float* __restrict__ qb, const float* __restrict__ offv,
                        const float* __restrict__ sclv,
                        int* __restrict__ rect, float* __restrict__ sval) {
    int k = threadIdx.x;                    // 256 threads, 1 block
    int q = k & 63;
    int cpr = (int)(qb[q * 2 + 0] * 128.0f);
    int cpc = (int)(qb[q * 2 + 1] * 128.0f);
    int cr = (int)((float)cpr + offv[k * 2 + 0] * 8.0f);   // trunc toward zero == astype(int32)
    int cc = (int)((float)cpc + offv[k * 2 + 1] * 8.0f);
    rect[k * 4 + 0] = cr - 8;
    rect[k * 4 + 1] = cr + 8;
    rect[k * 4 + 2] = cc - 8;
    rect[k * 4 + 3] = cc + 8;
    sval[k] = sclv[k];
}

// ---------------- Stage 7: scale_map (last covering query wins) ----------------
__global__ void scale_map_k(const int* __restrict__ rect, const float* __restrict__ sval,
                            float* __restrict__ smap) {
    int i = blockIdx.x * blockDim.x + threadIdx.x;   // 4*128*128
    if (i >= 4 * HW128) return;
    int w = i & 127, h = (i >> 7) & 127, b = i >> 14;
    float s = 1.0f;
    for (int q = 63; q >= 0; --q) {
        int k = (b << 6) + q;
        int rs = rect[k * 4 + 0], re = rect[k * 4 + 1];
        int cs = rect[k * 4 + 2], ce = rect[k * 4 + 3];
        if (rs >= 0 && h >= rs && h < re && cs >= 0 && w >= cs && w < ce) {
            s = sval[k];
            break;
        }
    }
    smap[i] = s;
}

// ---------------- Stage 8: fused output, 4 outputs/thread (b128 traffic) ----------------
__global__ void fuse_out_k(const float* __restrict__ a1, const float* __restrict__ a2,
                           const float* __restrict__ a3,
                           const float* __restrict__ n1, const float* __restrict__ n2,
                           const float* __restrict__ n3,
                           const float* __restrict__ smap, float* __restrict__ out) {
    int tid = blockIdx.x * blockDim.x + threadIdx.x;   // 50331648/4 = 12,582,912 threads
    int wq = tid & 31;
    int w0 = wq << 2;
    int t = tid >> 5;
    int h = t & 127;
    t >>= 7;
    int c = t % 768;
    int b = t / 768;

    float4 s4 = *(const float4*)(smap + (b << 14) + (h << 7) + w0);
    float4 f4, n4;
    if (c < 256) {
        size_t o = (((size_t)(b * 256 + c)) << 14) + (h << 7) + w0;
        f4 = *(const float4*)(a1 + o);
        n4 = *(const float4*)(n1 + o);
    } else if (c < 512) {
        size_t plane = (size_t)(b * 256 + (c - 256)) * HW64;
        f4 = bilin4(a2 + plane, 64, h, w0);
        n4 = bilin4(n2 + plane, 64, h, w0);
    } else {
        size_t plane = (size_t)(b * 256 + (c - 512)) * HW32;
        f4 = bilin4(a3 + plane, 32, h, w0);
        n4 = bilin4(n3 + plane, 32, h, w0);
    }
    float4 o4;
    o4.x = f4.x * s4.x + n4.x;
    o4.y = f4.y * s4.y + n4.y;
    o4.z = f4.z * s4.z + n4.z;
    o4.w = f4.w * s4.w + n4.w;
    *(float4*)(out + (size_t)tid * 4) = o4;
}

extern "C" void kernel_launch(void* const* d_in, const int* in_sizes, int n_in,
                              void* d_out, int out_size, void* d_ws, size_t ws_size,
                              hipStream_t stream) {
    (void)in_sizes; (void)n_in; (void)out_size; (void)ws_size;
    const float* t0_l1 = (const float*)d_in[1];
    const float* t0_l2 = (const float*)d_in[2];
    const float* t0_l3 = (const float*)d_in[3];
    const float* t1_l1 = (const float*)d_in[5];
    const float* t1_l2 = (const float*)d_in[6];
    const float* t1_l3 = (const float*)d_in[7];
    const float* qbbox = (const float*)d_in[8];
    const float* off_w1 = (const float*)d_in[9];
    const float* off_b1 = (const float*)d_in[10];
    const float* off_w2 = (const float*)d_in[11];
    const float* off_b2 = (const float*)d_in[12];
    const float* sc_w1 = (const float*)d_in[13];
    const float* sc_b1 = (const float*)d_in[14];
    const float* sc_w2 = (const float*)d_in[15];
    const float* sc_b2 = (const float*)d_in[16];
    float* out = (float*)d_out;

    // Workspace layout (floats); all offsets 16B-aligned
    float* ws    = (float*)d_ws;
    float* sumL1 = ws;                 // 65536
    float* sumL2 = sumL1 + 65536;      // 16384
    float* sumL3 = sumL2 + 16384;      // 4096
    float* meanf = sumL3 + 4096;       // 65536
    float* qf    = meanf + 65536;      // 65536
    float* Hoff  = qf    + 65536;      // 32768
    float* Hsc   = Hoff  + 32768;      // 32768
    float* offv  = Hsc   + 32768;      // 512
    float* sclv  = offv  + 512;        // 256
    int*   rect  = (int*)(sclv + 256); // 1024 ints
    float* sval  = (float*)(rect + 1024); // 256
    float* smap  = sval + 256;         // 65536

    const int TB = 256;
    channel_sum4_k<<<(4 * HW128 / 4 + TB - 1) / TB, TB, 0, stream>>>(t0_l1, sumL1, 256, HW128, 4 * HW128 / 4);
    channel_sum4_k<<<(4 * HW64  / 4 + TB - 1) / TB, TB, 0, stream>>>(t0_l2, sumL2, 256, HW64,  4 * HW64  / 4);
    channel_sum4_k<<<(4 * HW32  / 4 + TB - 1) / TB, TB, 0, stream>>>(t0_l3, sumL3, 256, HW32,  4 * HW32  / 4);
    mean_k<<<(4 * HW128 + TB - 1) / TB, TB, 0, stream>>>(sumL1, sumL2, sumL3, meanf);
    build_qf_k<<<(256 * 256 + TB - 1) / TB, TB, 0, stream>>>(meanf, qf);
    mlp_gemm1_k<<<dim3(16, 8, 2), 32, 0, stream>>>(qf, off_w1, off_b1, sc_w1, sc_b1, Hoff, Hsc);
    mlp_gemm2_k<<<dim3(16, 2, 1), 32, 0, stream>>>(Hoff, off_w2, off_b2, Hsc, sc_w2, sc_b2, offv, sclv);
    rects_k<<<1, 256, 0, stream>>>(qbbox, offv, sclv, rect, sval);
    scale_map_k<<<(4 * HW128 + TB - 1) / TB, TB, 0, stream>>>(rect, sval, smap);
    fuse_out_k<<<(4 * 768 * HW128 / 4) / TB, TB, 0, stream>>>(t0_l1, t0_l2, t0_l3,
                                                              t1_l1, t1_l2, t1_l3, smap, out);
}